// System2Reasoner_36670430773784
// MI455X (gfx1250) — compile-verified
//
#include <hip/hip_runtime.h>

typedef float v2f __attribute__((ext_vector_type(2)));
typedef float v8f __attribute__((ext_vector_type(8)));

#define P_ROWS 2048
#define DIM    768
#define MNODES 65536
#define TK     50
#define NS     8          // column slices
#define ROWS   32         // rows per block
#define COLS   64         // columns per step
#define KC     384        // K chunk staged in LDS (2 chunks per step)
#define NEG_BIG (-3.402823466e38f)

// LDS byte-offset layout for k_gemm_topk (dynamic LDS starts at offset 0)
#define OFF_AS    0u
#define OFF_BS0   (OFF_AS + (unsigned)(ROWS * DIM * 4))          //  98304
#define OFF_BS1   (OFF_BS0 + (unsigned)(COLS * KC * 4))          // 196608
#define OFF_SIMB  (OFF_BS1 + (unsigned)(COLS * KC * 4))          // 294912
#define OFF_LV    (OFF_SIMB + (unsigned)(ROWS * COLS * 4))       // 303104
#define OFF_LI    (OFF_LV + (unsigned)(ROWS * TK * 4))           // 309504
#define OFF_THR   (OFF_LI + (unsigned)(ROWS * TK * 4))           // 315904
#define LDS_TOTAL (OFF_THR + (unsigned)(ROWS * 4))               // 316032

// ---------------------------------------------------------------------------
// Kernel 1: fused f32 GEMM (WMMA 16x16x4) + running per-row top-50 per slice.
// B tiles are double-buffered in LDS via async global->LDS DMA (ASYNCcnt).
// ---------------------------------------------------------------------------
__global__ __launch_bounds__(256)
void k_gemm_topk(const float* __restrict__ X, const float* __restrict__ Mem,
                 float* __restrict__ pv, int* __restrict__ pi) {
  extern __shared__ char smem[];
  float* As   = (float*)(smem + OFF_AS);     // [32][768]
  float* Bs0  = (float*)(smem + OFF_BS0);    // [64][384]
  float* Bs1  = (float*)(smem + OFF_BS1);    // [64][384]
  float* simb = (float*)(smem + OFF_SIMB);   // [32][64]
  float* lv   = (float*)(smem + OFF_LV);     // [32][50]
  int*   li   = (int*)(smem + OFF_LI);       // [32][50]
  float* thr  = (float*)(smem + OFF_THR);    // [32]

  const int t    = threadIdx.x;
  const int lane = t & 31;
  const int wave = t >> 5;
  const int rt   = wave >> 2;                    // 0..1  row tile of 16
  const int ct   = wave & 3;                     // 0..3  col tile of 16
  const int rowBase  = blockIdx.x * ROWS;
  const int slice    = blockIdx.y;
  const int colStart = slice * (MNODES / NS);

  // Stage all 32 rows of X (contiguous) into LDS once.
  {
    const float4* src = (const float4*)(X + (size_t)rowBase * DIM);
    float4* dst = (float4*)As;
    for (int i = t; i < ROWS * DIM / 4; i += 256) dst[i] = src[i];
  }
  for (int i = t; i < ROWS * TK; i += 256) { lv[i] = NEG_BIG; li[i] = 0; }
  if (t < ROWS) thr[t] = NEG_BIG;
  __syncthreads();

  const int m0   = lane & 15;          // M (A) / N (B) within tile
  const int koff = (lane >> 4) * 2;    // K sub-offset per half-wave

  // Async-DMA one B chunk (64 cols x 384 K) into LDS buffer `buf`.
  // Each thread issues 24 x b128 copies: per-lane global addr -> per-lane LDS addr.
  auto issueB = [&](int buf, int kb, int colBase) {
    const unsigned bufBase = buf ? OFF_BS1 : OFF_BS0;
    const int F4PC = KC / 4;   // 96 float4 per column
#pragma unroll
    for (int i = 0; i < (COLS * KC / 4) / 256; ++i) {   // 24 iterations
      int f  = t + i * 256;
      int c  = f / F4PC;
      int kk = f % F4PC;
      unsigned ldsOff = bufBase + (unsigned)f * 16u;
      unsigned gOff   = (unsigned)((((size_t)(colBase + c)) * DIM + kb) * 4 +
                                   (size_t)kk * 16);
      asm volatile("global_load_async_to_lds_b128 %0, %1, %2"
                   :
                   : "v"(ldsOff), "v"(gOff), "s"(Mem)
                   : "memory");
    }
  };

  const int steps = (MNODES / NS) / COLS;   // 128
  for (int step = 0; step < steps; ++step) {
    const int colBase = colStart + step * COLS;
    issueB(0, 0, colBase);                  // prefetch first chunk
    v8f acc = {};

#pragma unroll
    for (int c = 0; c < DIM / KC; ++c) {    // 2 chunks
      asm volatile("s_wait_asynccnt 0x0" ::: "memory");  // our DMAs landed
      __syncthreads();                      // everyone's DMAs landed
      if (c == 0) issueB(1, KC, colBase);   // overlap: fill buf1 during compute

      const float* Ap = As + (size_t)(rt * 16 + m0) * DIM + c * KC + koff;
      const float* Bp = (c ? Bs1 : Bs0) + (size_t)(ct * 16 + m0) * KC + koff;
#pragma unroll 8
      for (int kk = 0; kk < KC; kk += 4) {
        v2f a = *(const v2f*)(Ap + kk);
        v2f b = *(const v2f*)(Bp + kk);
        acc = __builtin_amdgcn_wmma_f32_16x16x4_f32(
            /*neg_a=*/false, a, /*neg_b=*/false, b,
            /*c_mod=*/(short)0, acc, /*reuse_a=*/false, /*reuse_b=*/false);
      }
      __syncthreads();                      // chunk buffer free for reuse
    }

    // Dump this wave's 16x16 tile into the block sim buffer.
    {
      int n = lane & 15, hi = lane >> 4;
#pragma unroll
      for (int r = 0; r < 8; ++r)
        simb[(size_t)(rt * 16 + hi * 8 + r) * COLS + ct * 16 + n] = acc[r];
    }
    __syncthreads();

    // Threshold-guarded top-50 insertion: thread t owns row t (wave 0 only).
    if (t < ROWS) {
      float th = thr[t];
      for (int c = 0; c < COLS; ++c) {
        float v = simb[t * COLS + c];
        if (v > th) {
          int pos = 0; float mn = lv[t * TK];
          for (int j = 1; j < TK; ++j) {
            float x = lv[t * TK + j];
            if (x < mn) { mn = x; pos = j; }
          }
          lv[t * TK + pos] = v;
          li[t * TK + pos] = colBase + c;
          mn = lv[t * TK];
          for (int j = 1; j < TK; ++j) mn = fminf(mn, lv[t * TK + j]);
          th = mn;
        }
      }
      thr[t] = th;
    }
    __syncthreads();
  }

  // Write per-slice partial lists.
  for (int i = t; i < ROWS * TK; i += 256) {
    int r = i / TK, e = i % TK;
    size_t off = ((size_t)(rowBase + r) * NS + slice) * TK + e;
    pv[off] = lv[i];
    pi[off] = li[i];
  }
}

// ---------------------------------------------------------------------------
// Kernel 2: merge 8x50 partials -> final top-50 indices per row (desc order)
// ---------------------------------------------------------------------------
__global__ __launch_bounds__(256)
void k_merge(const float* __restrict__ pv, const int* __restrict__ pi,
             int* __restrict__ fidx) {
  const int row  = blockIdx.x * 8 + (threadIdx.x >> 5);
  const int lane = threadIdx.x & 31;
  const int NC = NS * TK;   // 400
  float cv[13]; int ci[13];
  int cnt = 0;
  for (int j = lane; j < NC; j += 32) {
    cv[cnt] = pv[(size_t)row * NC + j];
    ci[cnt] = pi[(size_t)row * NC + j];
    ++cnt;
  }
  for (int it = 0; it < TK; ++it) {
    float bv = NEG_BIG; int bi = 0x7fffffff; int bs = -1;
    for (int s = 0; s < cnt; ++s)
      if (cv[s] > bv || (cv[s] == bv && ci[s] < bi)) { bv = cv[s]; bi = ci[s]; bs = s; }
    float rv = bv; int ri = bi;
#pragma unroll
    for (int off = 16; off; off >>= 1) {
      float ov = __shfl_xor(rv, off, 32);
      int   oi = __shfl_xor(ri, off, 32);
      if (ov > rv || (ov == rv && oi < ri)) { rv = ov; ri = oi; }
    }
    if (lane == 0) fidx[(size_t)row * TK + it] = ri;
    if (bs >= 0 && ci[bs] == ri) cv[bs] = NEG_BIG;   // winner retires its slot
  }
}

// ---------------------------------------------------------------------------
// unique via flags + exclusive scan + scatter (reproduces jnp.unique exactly)
// ---------------------------------------------------------------------------
__global__ void k_init(int* __restrict__ flags, int* __restrict__ uniq) {
  int i = blockIdx.x * blockDim.x + threadIdx.x;
  if (i < P_ROWS * TK) uniq[i] = 0;          // fill_value = 0 padding
  if (i < MNODES) flags[i] = 0;
}

__global__ void k_mark(const int* __restrict__ fidx, int* __restrict__ flags) {
  int i = blockIdx.x * blockDim.x + threadIdx.x;
  if (i < P_ROWS * TK) flags[fidx[i]] = 1;   // all writers store 1: race-free
}

__global__ __launch_bounds__(1024)
void k_scan(const int* __restrict__ flags, int* __restrict__ rank,
            int* __restrict__ ucount) {
  __shared__ int part[1024];
  const int t = threadIdx.x;
  const int base = t * (MNODES / 1024);   // 64 per thread
  int s = 0;
  for (int i = 0; i < MNODES / 1024; ++i) s += flags[base + i];
  part[t] = s;
  __syncthreads();
  for (int off = 1; off < 1024; off <<= 1) {
    int v = part[t];
    int add = (t >= off) ? part[t - off] : 0;
    __syncthreads();
    part[t] = v + add;
    __syncthreads();
  }
  int run = part[t] - s;                  // exclusive
  for (int i = 0; i < MNODES / 1024; ++i) {
    rank[base + i] = run;
    run += flags[base + i];
  }
  if (t == 1023) *ucount = part[1023];
}

__global__ void k_scatter(const int* __restrict__ flags, const int* __restrict__ rank,
                          int* __restrict__ uniq) {
  int v = blockIdx.x * blockDim.x + threadIdx.x;
  if (v < MNODES && flags[v]) uniq[rank[v]] = v;   // ascending-sorted uniques
}

__global__ void k_edges(const int* __restrict__ fidx, const int* __restrict__ rank,
                        int* __restrict__ edge_out) {
  int i = blockIdx.x * blockDim.x + threadIdx.x;
  if (i < P_ROWS * TK) {
    edge_out[i]               = rank[fidx[i]];   // inverse mapping
    edge_out[P_ROWS * TK + i] = i / TK;          // test_node_idx
  }
}

// ---------------------------------------------------------------------------
// active_memory_nodes = memory_nodes[uniq]  (315 MB coalesced gather)
// ---------------------------------------------------------------------------
__global__ __launch_bounds__(192)
void k_gather(const float* __restrict__ Mem, const int* __restrict__ uniq,
              float* __restrict__ out) {
  const int j = blockIdx.x;
  const int src = uniq[j];
  const float4* s = (const float4*)(Mem + (size_t)src * DIM);
  float4* d = (float4*)(out + (size_t)j * DIM);
  d[threadIdx.x] = s[threadIdx.x];   // 192 * float4 = 768 floats
}

// ---------------------------------------------------------------------------
// evidence MLP logits: relu(X@W1 + b1) @ W2 + b2
// ---------------------------------------------------------------------------
__global__ __launch_bounds__(256)
void k_mlp(const float* __restrict__ X, const float* __restrict__ W1,
           const float* __restrict__ b1, const float* __restrict__ W2,
           const float* __restrict__ b2, float* __restrict__ logits) {
  __shared__ float xs[DIM];
  __shared__ float red[256];
  const int row = blockIdx.x, t = threadIdx.x;
  for (int i = t; i < DIM; i += 256) xs[i] = X[(size_t)row * DIM + i];
  __syncthreads();
  float local = 0.f;
  for (int j = t; j < DIM / 2; j += 256) {
    float acc = b1[j];
    for (int d = 0; d < DIM; ++d) acc = fmaf(xs[d], W1[(size_t)d * (DIM / 2) + j], acc);
    local += fmaxf(acc, 0.f) * W2[j];
  }
  red[t] = local;
  __syncthreads();
  for (int off = 128; off; off >>= 1) {
    if (t < off) red[t] += red[t + off];
    __syncthreads();
  }
  if (t == 0) logits[row] = red[0] + b2[0];
}

// ---------------------------------------------------------------------------
// softmax over P + weighted pooling + L2 normalize (single block)
// ---------------------------------------------------------------------------
__global__ __launch_bounds__(1024)
void k_pool(const float* __restrict__ X, const float* __restrict__ logits,
            float* __restrict__ out_gf) {
  __shared__ float red[1024];
  __shared__ float gf[DIM];
  __shared__ float mx_s, sum_s;
  const int t = threadIdx.x;

  float m = NEG_BIG;
  for (int p = t; p < P_ROWS; p += 1024) m = fmaxf(m, logits[p]);
  red[t] = m; __syncthreads();
  for (int off = 512; off; off >>= 1) { if (t < off) red[t] = fmaxf(red[t], red[t + off]); __syncthreads(); }
  if (t == 0) mx_s = red[0];
  __syncthreads();
  const float mx = mx_s;

  float s = 0.f;
  for (int p = t; p < P_ROWS; p += 1024) s += expf(logits[p] - mx);
  red[t] = s; __syncthreads();
  for (int off = 512; off; off >>= 1) { if (t < off) red[t] += red[t + off]; __syncthreads(); }
  if (t == 0) sum_s = red[0];
  __syncthreads();
  const float inv = 1.f / sum_s;

  if (t < DIM) {
    float a = 0.f;
    for (int p = 0; p < P_ROWS; ++p)
      a = fmaf(expf(logits[p] - mx) * inv, X[(size_t)p * DIM + t], a);
    gf[t] = a;
  }
  __syncthreads();

  red[t] = (t < DIM) ? gf[t] * gf[t] : 0.f;
  __syncthreads();
  for (int off = 512; off; off >>= 1) { if (t < off) red[t] += red[t + off]; __syncthreads(); }
  if (t == 0) sum_s = fmaxf(sqrtf(red[0]), 1e-12f);
  __syncthreads();
  if (t < DIM) out_gf[t] = gf[t] / sum_s;
}

// ---------------------------------------------------------------------------
extern "C" void kernel_launch(void* const* d_in, const int* in_sizes, int n_in,
                              void* d_out, int out_size, void* d_ws, size_t ws_size,
                              hipStream_t stream) {
  const float* X   = (const float*)d_in[0];   // [2048, 768]
  const float* Mem = (const float*)d_in[1];   // [65536, 768]
  const float* W1  = (const float*)d_in[2];   // [768, 384]
  const float* b1  = (const float*)d_in[3];   // [384]
  const float* W2  = (const float*)d_in[4];   // [384, 1]
  const float* b2  = (const float*)d_in[5];   // [1]

  const int PK = P_ROWS * TK;                 // 102400

  // workspace partition
  char* w = (char*)d_ws;
  float* pv   = (float*)w;               w += (size_t)P_ROWS * NS * TK * 4;
  int*   pi   = (int*)w;                 w += (size_t)P_ROWS * NS * TK * 4;
  int*   fidx = (int*)w;                 w += (size_t)PK * 4;
  int*   flags= (int*)w;                 w += (size_t)MNODES * 4;
  int*   rank = (int*)w;                 w += (size_t)MNODES * 4;
  int*   uniq = (int*)w;                 w += (size_t)PK * 4;
  float* logits = (float*)w;             w += (size_t)P_ROWS * 4;
  int*   ucount = (int*)w;               w += 256;

  // output partition: [2,102400] int edges | [102400,768] f32 | [768] f32
  int*   edge_out   = (int*)d_out;
  float* active_out = (float*)d_out + 2 * PK;
  float* gf_out     = (float*)d_out + 2 * PK + (size_t)PK * DIM;

  // 1) fused GEMM (WMMA f32 16x16x4, async double-buffered B) + per-slice top-50
  k_gemm_topk<<<dim3(P_ROWS / ROWS, NS), 256, LDS_TOTAL, stream>>>(X, Mem, pv, pi);
  // 2) merge partials -> final top-50 indices
  k_merge<<<P_ROWS / 8, 256, 0, stream>>>(pv, pi, fidx);
  // 3) unique = flags + scan + scatter
  k_init<<<(PK + 255) / 256, 256, 0, stream>>>(flags, uniq);
  k_mark<<<(PK + 255) / 256, 256, 0, stream>>>(fidx, flags);
  k_scan<<<1, 1024, 0, stream>>>(flags, rank, ucount);
  k_scatter<<<(MNODES + 255) / 256, 256, 0, stream>>>(flags, rank, uniq);
  k_edges<<<(PK + 255) / 256, 256, 0, stream>>>(fidx, rank, edge_out);
  // 4) gather active memory nodes
  k_gather<<<PK, 192, 0, stream>>>(Mem, uniq, active_out);
  // 5) evidence pooling
  k_mlp<<<P_ROWS, 256, 0, stream>>>(X, W1, b1, W2, b2, logits);
  k_pool<<<1, 1024, 0, stream>>>(X, logits, gf_out);
}